// TTLoRALinearWrapper_Reconstruction_13030930776376
// MI455X (gfx1250) — compile-verified
//
#include <hip/hip_runtime.h>
#include <hip/hip_bf16.h>

// ---------------------------------------------------------------------------
// TT-LoRA linear:  out = x @ (W + 8*TT)^T + bias
//   x: (4,2048,2048) f32  -> M=8192, K=2048
//   W: (2048,2048)  f32   -> N=2048 rows (output features), K cols
// Strategy:
//   1) tt_factors: L(2048x8) = c0*c1*c2*c3 ; R(8x2048) = c4*c5*c6*c7
//   2) build_adapted: A = W + 8*(L@R)  (workspace, 16 MB; L2-resident)
//   3) tt_gemm: fp32-exact GEMM on the CDNA5 matrix pipe via
//      V_WMMA_F32_16X16X4_F32. Block tile 128x64, wave tile 16x64
//      (4 accumulators -> 1.25 LDS loads per WMMA), LDS fed by
//      double-buffered global_load_async_to_lds_b128 (ASYNCcnt).
// ---------------------------------------------------------------------------

typedef float v2f __attribute__((ext_vector_type(2)));
typedef float v8f __attribute__((ext_vector_type(8)));

#define DMODEL 2048
#define TILE_K 32
#define LSTR   36   // padded floats per LDS row (144 B: keeps b128 writes 16B-aligned)

// ---------------------------------------------------------------------------
// Kernel 1: build L (2048x8) and R (8x2048) from the TT cores.
// row = ((j*8+l)*8+n)*4+p ; col = ((r*8+t)*8+v)*8+x
// ---------------------------------------------------------------------------
__global__ __launch_bounds__(256) void tt_factors(
    const float* __restrict__ c0, const float* __restrict__ c1,
    const float* __restrict__ c2, const float* __restrict__ c3,
    const float* __restrict__ c4, const float* __restrict__ c5,
    const float* __restrict__ c6, const float* __restrict__ c7,
    float* __restrict__ L, float* __restrict__ R)
{
  const int idx = blockIdx.x * 256 + threadIdx.x;
  float v[8], w[8];
  if (idx < 2048) {
    const int row = idx;
    const int p = row & 3, n = (row >> 2) & 7, l = (row >> 5) & 7, j = row >> 8;
    #pragma unroll
    for (int s = 0; s < 8; ++s) v[s] = c0[j * 8 + s];                 // (1,8,8)
    #pragma unroll
    for (int s2 = 0; s2 < 8; ++s2) {                                  // c1 (8,8,8)
      float a = 0.f;
      #pragma unroll
      for (int s = 0; s < 8; ++s) a += v[s] * c1[(s * 8 + l) * 8 + s2];
      w[s2] = a;
    }
    #pragma unroll
    for (int s2 = 0; s2 < 8; ++s2) {                                  // c2 (8,8,8)
      float a = 0.f;
      #pragma unroll
      for (int s = 0; s < 8; ++s) a += w[s] * c2[(s * 8 + n) * 8 + s2];
      v[s2] = a;
    }
    #pragma unroll
    for (int s2 = 0; s2 < 8; ++s2) {                                  // c3 (8,4,8)
      float a = 0.f;
      #pragma unroll
      for (int s = 0; s < 8; ++s) a += v[s] * c3[(s * 4 + p) * 8 + s2];
      w[s2] = a;
    }
    #pragma unroll
    for (int s2 = 0; s2 < 8; ++s2) L[row * 8 + s2] = w[s2];
  } else if (idx < 4096) {
    const int col = idx - 2048;
    const int x = col & 7, vv = (col >> 3) & 7, tq = (col >> 6) & 7, r = col >> 9;
    #pragma unroll
    for (int d = 0; d < 8; ++d) v[d] = c7[d * 8 + x];                 // (8,8,1)
    #pragma unroll
    for (int c = 0; c < 8; ++c) {                                     // c6 (8,8,8)
      float a = 0.f;
      #pragma unroll
      for (int d = 0; d < 8; ++d) a += c6[(c * 8 + vv) * 8 + d] * v[d];
      w[c] = a;
    }
    #pragma unroll
    for (int b = 0; b < 8; ++b) {                                     // c5 (8,8,8)
      float a = 0.f;
      #pragma unroll
      for (int c = 0; c < 8; ++c) a += c5[(b * 8 + tq) * 8 + c] * w[c];
      v[b] = a;
    }
    #pragma unroll
    for (int aa = 0; aa < 8; ++aa) {                                  // c4 (8,4,8)
      float a = 0.f;
      #pragma unroll
      for (int b = 0; b < 8; ++b) a += c4[(aa * 4 + r) * 8 + b] * v[b];
      R[aa * DMODEL + col] = a;
    }
  }
}

// ---------------------------------------------------------------------------
// Kernel 2: A = W + 8 * (L @ R)   (one float4 per thread, 1M threads)
// ---------------------------------------------------------------------------
__global__ __launch_bounds__(256) void build_adapted(
    const float* __restrict__ W, const float* __restrict__ L,
    const float* __restrict__ R, float* __restrict__ A)
{
  const int idx = blockIdx.x * 256 + threadIdx.x;   // float4 index in [0, 1M)
  const int row = idx >> 9;                         // 512 float4 per 2048-row
  const int c0  = (idx & 511) << 2;
  const float* wp = W + (size_t)row * DMODEL + c0;
  float a0 = 0.f, a1 = 0.f, a2 = 0.f, a3 = 0.f;
  #pragma unroll
  for (int s = 0; s < 8; ++s) {
    const float ls = L[row * 8 + s];
    const float* rp = R + s * DMODEL + c0;
    a0 += ls * rp[0]; a1 += ls * rp[1]; a2 += ls * rp[2]; a3 += ls * rp[3];
  }
  float* ap = A + (size_t)row * DMODEL + c0;
  ap[0] = wp[0] + 8.0f * a0;
  ap[1] = wp[1] + 8.0f * a1;
  ap[2] = wp[2] + 8.0f * a2;
  ap[3] = wp[3] + 8.0f * a3;
}

// ---------------------------------------------------------------------------
// Kernel 3: out[m,n] = sum_k X[m,k]*A[n,k] + bias[n]
//   fp32 WMMA 16x16x4. Block tile 128(M)x64(N), 8 waves, wave tile 16x64.
//   Double-buffered async global->LDS copies (6 async instrs/wave/buffer).
// ---------------------------------------------------------------------------
__device__ __forceinline__ void async_b128(unsigned lds_off, const float* gp) {
  asm volatile("global_load_async_to_lds_b128 %0, %1, off"
               :: "v"(lds_off), "v"((unsigned long long)(size_t)gp)
               : "memory");
}

__global__ __launch_bounds__(256) void tt_gemm(
    const float* __restrict__ X, const float* __restrict__ A,
    const float* __restrict__ bias, float* __restrict__ out,
    int M, int N, int K)
{
  __shared__ float Xs[2][128 * LSTR];   // 128 rows of x, 32-K window
  __shared__ float Ws[2][64 * LSTR];    // 64 rows of adapted W, 32-K window

  const int t    = threadIdx.x;
  const int lane = t & 31;
  const int wave = t >> 5;              // 0..7 -> 16-row slab in M
  const int ln   = lane & 15;           // M (A-frag) / N (B-frag) index
  const int hi   = lane >> 4;           // K-half select per ISA fragment layout

  const int bm = blockIdx.y * 128;
  const int bn = blockIdx.x * 64;

  const float* Xg = X + (size_t)bm * K;   // 128 rows of x
  const float* Ag = A + (size_t)bn * K;   // 64 rows of adapted weight

  // Stage a 32-float-wide K window: X tile = 1024 16B chunks (4/thread),
  // W tile = 512 chunks (2/thread).
  auto issue = [&](int buf, int k0) {
    const unsigned xbase = (unsigned)(size_t)&Xs[buf][0];
    const unsigned wbase = (unsigned)(size_t)&Ws[buf][0];
    #pragma unroll
    for (int i = 0; i < 4; ++i) {
      const int c = i * 256 + t;
      const int row = c >> 3, sub = (c & 7) * 4;
      async_b128(xbase + (unsigned)(row * LSTR + sub) * 4u,
                 Xg + (size_t)row * K + k0 + sub);
    }
    #pragma unroll
    for (int i = 0; i < 2; ++i) {
      const int c = i * 256 + t;
      const int row = c >> 3, sub = (c & 7) * 4;
      async_b128(wbase + (unsigned)(row * LSTR + sub) * 4u,
                 Ag + (size_t)row * K + k0 + sub);
    }
  };

  v8f acc[4];
  #pragma unroll
  for (int j = 0; j < 4; ++j) acc[j] = (v8f){0.f,0.f,0.f,0.f,0.f,0.f,0.f,0.f};

  // A-frag: lane ln holds M=wave*16+ln, K pair (2*hi + {0,1})
  // B-frag: lane ln holds N=j*16+ln, same K pairing (contiguous float2)
  auto compute = [&](int buf) {
    const float* xs  = &Xs[buf][(wave * 16 + ln) * LSTR + 2 * hi];
    const float* wsb = &Ws[buf][ln * LSTR + 2 * hi];
    #pragma unroll
    for (int kq = 0; kq < TILE_K; kq += 4) {
      v2f a = *(const v2f*)(xs + kq);
      #pragma unroll
      for (int j = 0; j < 4; ++j) {
        v2f b = *(const v2f*)(wsb + j * 16 * LSTR + kq);
        acc[j] = __builtin_amdgcn_wmma_f32_16x16x4_f32(
            false, a, false, b, (short)0, acc[j], false, false);
      }
    }
  };

  const int ksteps = K / TILE_K;
  issue(0, 0);                                   // prefetch buffer 0
  for (int ks = 0; ks < ksteps - 1; ++ks) {
    issue((ks + 1) & 1, (ks + 1) * TILE_K);      // 12 outstanding
    asm volatile("s_wait_asynccnt 6" ::: "memory");  // current buffer landed
    __syncthreads();
    compute(ks & 1);
    __syncthreads();                             // buf free before refill
  }
  asm volatile("s_wait_asynccnt 0" ::: "memory");
  __syncthreads();
  compute((ksteps - 1) & 1);

  // C/D layout: VGPR v -> M = v + 8*hi, N = ln
  const int row0 = bm + wave * 16 + hi * 8;
  #pragma unroll
  for (int j = 0; j < 4; ++j) {
    const int col = bn + j * 16 + ln;
    const float bv = bias[col];
    #pragma unroll
    for (int v = 0; v < 8; ++v)
      out[(size_t)(row0 + v) * N + col] = acc[j][v] + bv;
  }
}

// ---------------------------------------------------------------------------
extern "C" void kernel_launch(void* const* d_in, const int* in_sizes, int n_in,
                              void* d_out, int out_size, void* d_ws, size_t ws_size,
                              hipStream_t stream) {
  const float* x    = (const float*)d_in[0];
  const float* W    = (const float*)d_in[1];
  const float* bias = (const float*)d_in[2];
  const float* c0 = (const float*)d_in[3];
  const float* c1 = (const float*)d_in[4];
  const float* c2 = (const float*)d_in[5];
  const float* c3 = (const float*)d_in[6];
  const float* c4 = (const float*)d_in[7];
  const float* c5 = (const float*)d_in[8];
  const float* c6 = (const float*)d_in[9];
  const float* c7 = (const float*)d_in[10];

  float* L = (float*)d_ws;                 // 2048 x 8
  float* R = L + 2048 * 8;                 // 8 x 2048
  float* A = R + 8 * 2048;                 // 2048 x 2048 adapted weight
  float* out = (float*)d_out;

  tt_factors<<<16, 256, 0, stream>>>(c0, c1, c2, c3, c4, c5, c6, c7, L, R);
  build_adapted<<<(DMODEL * DMODEL / 4) / 256, 256, 0, stream>>>(W, L, R, A);

  const int M = 4 * 2048, N = DMODEL, K = DMODEL;
  dim3 grid(N / 64, M / 128);
  tt_gemm<<<grid, dim3(256), 0, stream>>>(x, A, bias, out, M, N, K);
}